// EdgeFeatures_6339371729570
// MI455X (gfx1250) — compile-verified
//
#include <hip/hip_runtime.h>
#include <hip/hip_bf16.h>

typedef float v2f __attribute__((ext_vector_type(2)));
typedef float v8f __attribute__((ext_vector_type(8)));

#define BB   8
#define NN   8192
#define KK   48
#define NE   (BB * NN * KK)   // 3,145,728 edges
#define NF   23               // feature dim
#define NFP  25               // padded LDS row (odd stride -> conflict free)
#define NOUT 16
#define NFREQ 8
#define EPSV 1e-12f

__device__ __forceinline__ float3 loadf3(const float* __restrict__ p, int idx) {
    return make_float3(p[idx * 3 + 0], p[idx * 3 + 1], p[idx * 3 + 2]);
}
__device__ __forceinline__ float3 sub3(float3 a, float3 b) {
    return make_float3(a.x - b.x, a.y - b.y, a.z - b.z);
}
__device__ __forceinline__ float3 cross3(float3 a, float3 b) {
    return make_float3(a.y * b.z - a.z * b.y,
                       a.z * b.x - a.x * b.z,
                       a.x * b.y - a.y * b.x);
}
__device__ __forceinline__ float3 norm3(float3 a) {
    float n   = sqrtf(a.x * a.x + a.y * a.y + a.z * a.z);
    float inv = 1.0f / fmaxf(n, EPSV);
    return make_float3(a.x * inv, a.y * inv, a.z * inv);
}
__device__ __forceinline__ float signf(float x) {
    return (x > 0.f) ? 1.f : ((x < 0.f) ? -1.f : 0.f);
}

// ---------------------------------------------------------------------------
// Kernel A: per-node orientation frames O[b,n] (row-major 3x3 = 9 floats).
// Frame valid for n in [1, N-3]; zeros elsewhere (matches jnp.pad((1,2))).
// ---------------------------------------------------------------------------
__global__ void __launch_bounds__(256) frames_kernel(const float* __restrict__ p,
                                                     float* __restrict__ O) {
    int idx = blockIdx.x * 256 + threadIdx.x;   // b*N + n
    if (idx >= BB * NN) return;
    int n = idx & (NN - 1);
    float o[9];
#pragma unroll
    for (int i = 0; i < 9; i++) o[i] = 0.f;
    if (n >= 1 && n <= NN - 3) {
        float3 xm = loadf3(p, idx - 1);
        float3 x0 = loadf3(p, idx);
        float3 xp = loadf3(p, idx + 1);
        float3 u2 = norm3(sub3(x0, xm));     // U[n-1]
        float3 u1 = norm3(sub3(xp, x0));     // U[n]
        float3 o1 = norm3(sub3(u2, u1));
        float3 n2 = norm3(cross3(u2, u1));
        float3 c3 = cross3(o1, n2);          // not normalized (matches reference)
        o[0] = o1.x; o[1] = o1.y; o[2] = o1.z;
        o[3] = n2.x; o[4] = n2.y; o[5] = n2.z;
        o[6] = c3.x; o[7] = c3.y; o[8] = c3.z;
    }
#pragma unroll
    for (int i = 0; i < 9; i++) O[idx * 9 + i] = o[i];
}

// ---------------------------------------------------------------------------
// Kernel B: per-edge features (23) -> LDS -> WMMA f32 16x16x4 projection to 16
// outputs, + bias, store h_e (un-normalized) to out.
// ---------------------------------------------------------------------------
__global__ void __launch_bounds__(256) edge_kernel(
    const float* __restrict__ p, const int* __restrict__ eidx,
    const float* __restrict__ Ofr,
    const float* __restrict__ tfrq, const float* __restrict__ ts,
    const float* __restrict__ W, const float* __restrict__ bias,
    float* __restrict__ out)
{
    __shared__ float feat[8][32][NFP];   // [wave][edge-in-wave][feature]

    const int tid  = threadIdx.x;
    const int wave = tid >> 5;
    const int lane = tid & 31;
    const int e    = blockIdx.x * 256 + tid;      // global edge id

    const int b    = e / (NN * KK);
    const int rem  = e - b * (NN * KK);
    const int n    = rem / KK;
    const int node = b * NN + n;
    const int j    = eidx[e];
    const int nodeJ = b * NN + j;

    // --- geometry features ---------------------------------------------------
    float Fo[9], Fj[9];
#pragma unroll
    for (int i = 0; i < 9; i++) { Fo[i] = Ofr[node * 9 + i]; Fj[i] = Ofr[nodeJ * 9 + i]; }

    float3 xn = loadf3(p, node);
    float3 xj = loadf3(p, nodeJ);
    float3 dx = sub3(xj, xn);

    // dU = l2norm(Of @ dX)
    float3 t;
    t.x = Fo[0] * dx.x + Fo[1] * dx.y + Fo[2] * dx.z;
    t.y = Fo[3] * dx.x + Fo[4] * dx.y + Fo[5] * dx.z;
    t.z = Fo[6] * dx.x + Fo[7] * dx.y + Fo[8] * dx.z;
    float3 dU = norm3(t);

    // R = Of^T @ O_nb :  R[i][l] = sum_m Fo[m*3+i] * Fj[m*3+l]
    float R[3][3];
#pragma unroll
    for (int i = 0; i < 3; i++)
#pragma unroll
        for (int l = 0; l < 3; l++)
            R[i][l] = Fo[i] * Fj[l] + Fo[3 + i] * Fj[3 + l] + Fo[6 + i] * Fj[6 + l];

    float Rxx = R[0][0], Ryy = R[1][1], Rzz = R[2][2];
    float m0 = 0.5f * sqrtf(fabsf(1.f + Rxx - Ryy - Rzz) + EPSV);
    float m1 = 0.5f * sqrtf(fabsf(1.f - Rxx + Ryy - Rzz) + EPSV);
    float m2 = 0.5f * sqrtf(fabsf(1.f - Rxx - Ryy + Rzz) + EPSV);
    float qx = signf(R[2][1] - R[1][2]) * m0;
    float qy = signf(R[0][2] - R[2][0]) * m1;
    float qz = signf(R[1][0] - R[0][1]) * m2;
    float qw = sqrtf(fmaxf(1.f + Rxx + Ryy + Rzz, 0.f) + EPSV) * 0.5f;
    float qn  = sqrtf(qx * qx + qy * qy + qz * qz + qw * qw);
    float qiv = 1.0f / fmaxf(qn, EPSV);
    qx *= qiv; qy *= qiv; qz *= qiv; qw *= qiv;

    // --- lrpe features -------------------------------------------------------
    float d   = (float)j - (float)n;
    float tss = ts[0];
    float* fp = &feat[wave][lane][0];
#pragma unroll
    for (int f = 0; f < NFREQ; f++) {
        float ang = d * (tfrq[f] * tss);
        fp[f]         = sinf(ang);
        fp[NFREQ + f] = cosf(ang);
    }
    fp[16] = dU.x; fp[17] = dU.y; fp[18] = dU.z;
    fp[19] = qx;   fp[20] = qy;   fp[21] = qz;   fp[22] = qw;
    fp[23] = 0.f;  // K-pad so f=23 contributes zero

    __syncthreads();

    // --- WMMA projection: (32 edges x 24) @ (24 x 16), f32 16x16x4 -----------
    const int col  = lane & 15;
    const int half = lane >> 4;

    // B fragment (W^T chunks): lanes 0-15 hold K=f0,f0+1; lanes 16-31 K=f0+2,f0+3
    v2f bf[6];
#pragma unroll
    for (int kc = 0; kc < 6; kc++) {
        int f0 = kc * 4 + 2 * half;
        bf[kc].x = (f0     < NF) ? W[col * NF + f0]     : 0.f;
        bf[kc].y = (f0 + 1 < NF) ? W[col * NF + f0 + 1] : 0.f;
    }
    float bb = bias[col];

#pragma unroll
    for (int tile = 0; tile < 2; tile++) {
        v8f c = {};
#pragma unroll
        for (int kc = 0; kc < 6; kc++) {
            const float* src = &feat[wave][tile * 16 + col][kc * 4 + 2 * half];
            v2f a;
            a.x = src[0];
            a.y = src[1];
            // D = A(16x4) x B(4x16) + C  (f32, exact vs reference)
            c = __builtin_amdgcn_wmma_f32_16x16x4_f32(
                    /*neg_a=*/false, a, /*neg_b=*/false, bf[kc],
                    /*c_mod=*/(short)0, c, /*reuse_a=*/false, /*reuse_b=*/false);
        }
        // C/D layout: VGPR r holds row r + 8*half, column = lane&15
        int rowBase = blockIdx.x * 256 + wave * 32 + tile * 16 + 8 * half;
#pragma unroll
        for (int r = 0; r < 8; r++) {
            out[(rowBase + r) * NOUT + col] = c[r] + bb;
        }
    }
}

// ---------------------------------------------------------------------------
// Kernel C: per-(b,k,o) inverse norms over the N axis (8192 terms).
// One block per (b,k): 16 n-chunks x 16 outputs, LDS reduction.
// ---------------------------------------------------------------------------
__global__ void __launch_bounds__(256) norm_kernel(const float* __restrict__ out,
                                                   float* __restrict__ scales) {
    __shared__ float red[256];
    const int bk    = blockIdx.x;          // b*KK + k
    const int b     = bk / KK;
    const int k     = bk - b * KK;
    const int tid   = threadIdx.x;
    const int o     = tid & 15;
    const int chunk = tid >> 4;            // 16 chunks of 512 n's

    const float* base = out + ((size_t)(b * NN) * KK + k) * NOUT + o;
    float s = 0.f;
    const int n0 = chunk * 512;
    for (int i = 0; i < 512; i++) {
        float v = base[(size_t)(n0 + i) * (KK * NOUT)];
        s += v * v;
    }
    red[tid] = s;
    __syncthreads();
    if (chunk == 0) {
        float tot = 0.f;
#pragma unroll
        for (int c = 0; c < 16; c++) tot += red[o + 16 * c];
        scales[bk * NOUT + o] = 1.0f / fmaxf(sqrtf(tot), EPSV);
    }
}

// ---------------------------------------------------------------------------
// Kernel D: in-place scale, float4 vectorized (o is the fastest axis and
// 4 | 16, so each float4 stays inside one (b,k) scale row).
// ---------------------------------------------------------------------------
__global__ void __launch_bounds__(256) scale_kernel(float* __restrict__ out,
                                                    const float* __restrict__ scales) {
    const int total4 = NE * NOUT / 4;
    int idx = blockIdx.x * 256 + threadIdx.x;
    if (idx >= total4) return;
    int i4 = idx * 4;
    int b  = i4 / (NN * KK * NOUT);
    int r  = i4 % (KK * NOUT);
    float4 v = ((const float4*)out)[idx];
    const float* sp = scales + b * (KK * NOUT) + r;
    v.x *= sp[0]; v.y *= sp[1]; v.z *= sp[2]; v.w *= sp[3];
    ((float4*)out)[idx] = v;
}

// ---------------------------------------------------------------------------
extern "C" void kernel_launch(void* const* d_in, const int* in_sizes, int n_in,
                              void* d_out, int out_size, void* d_ws, size_t ws_size,
                              hipStream_t stream) {
    const float* p    = (const float*)d_in[0];   // (B,N,3)
    const int*   eidx = (const int*)  d_in[1];   // (B,N,K) int32
    // d_in[2] = mask (unused by the reference math)
    const float* tfrq = (const float*)d_in[3];   // (8,)
    const float* ts   = (const float*)d_in[4];   // scalar
    const float* W    = (const float*)d_in[5];   // (16,23)
    const float* bias = (const float*)d_in[6];   // (16,)
    float* out = (float*)d_out;                  // (B,N,K,16) f32

    float* Ofr    = (float*)d_ws;                        // B*N*9 floats (2.4 MB)
    float* scales = Ofr + (size_t)BB * NN * 9;           // B*K*16 floats

    frames_kernel<<<(BB * NN) / 256, 256, 0, stream>>>(p, Ofr);
    edge_kernel<<<NE / 256, 256, 0, stream>>>(p, eidx, Ofr, tfrq, ts, W, bias, out);
    norm_kernel<<<BB * KK, 256, 0, stream>>>(out, scales);
    scale_kernel<<<(NE * NOUT / 4) / 256, 256, 0, stream>>>(out, scales);
}